// GraphSAGELayer_55963423867334
// MI455X (gfx1250) — compile-verified
//
#include <hip/hip_runtime.h>

#define NNODES 100000
#define DFEAT  64
#define DOUT   64
#define NODES_PER_BLOCK 32

typedef float v2f __attribute__((ext_vector_type(2)));
typedef float v8f __attribute__((ext_vector_type(8)));

// ---------------- Kernel 1: zero workspace (neigh_sum + deg) ----------------
__global__ void __launch_bounds__(256) sage_zero_ws(float* __restrict__ ws, long long n) {
  long long i = (long long)blockIdx.x * blockDim.x + threadIdx.x;
  long long stride = (long long)gridDim.x * blockDim.x;
  for (; i < n; i += stride) ws[i] = 0.0f;
}

// ---------------- Kernel 2: edge scatter-add (mean-agg numerator + degree) --
// One thread per (edge, feature). All 32 lanes of a wave touch <=1 edge's
// feature row halves -> src/dst loads are wave-uniform broadcasts; the
// feature gather/atomic is fully coalesced. f32 atomics resolve in L2
// (x and neigh_sum are ~25.6MB each, L2-resident on a 192MB L2).
__global__ void __launch_bounds__(256)
sage_scatter(const float* __restrict__ x,
             const int*   __restrict__ edge_src,
             const int*   __restrict__ edge_dst,
             float* __restrict__ neigh_sum,
             float* __restrict__ deg,
             long long nwork /* E * 64 */) {
  long long i = (long long)blockIdx.x * blockDim.x + threadIdx.x;
  long long stride = (long long)gridDim.x * blockDim.x;
  for (; i < nwork; i += stride) {
    long long e = i >> 6;           // edge id
    int f = (int)(i & 63);          // feature id
    int s = edge_src[e];
    int d = edge_dst[e];
    float v = x[(long long)s * DFEAT + f];
    atomicAdd(&neigh_sum[(long long)d * DFEAT + f], v);
    if (f == 0) atomicAdd(&deg[d], 1.0f);
  }
}

// ---------------- Kernel 3: normalize + concat + WMMA GEMM + bias ----------
// Each block: 32 nodes. LDS holds combined[32][128] tile and the full W
// [128][64] (padded). 8 waves -> wave w computes the 16x16 output tile
// (mt = w/4 rows, nt = w%4 cols) via 32 chained V_WMMA_F32_16X16X4_F32.
__global__ void __launch_bounds__(256)
sage_gemm(const float* __restrict__ x,
          const float* __restrict__ W,      // [128][64] row-major
          const float* __restrict__ bias,   // [64]
          const float* __restrict__ neigh_sum,
          const float* __restrict__ deg,
          float* __restrict__ out) {        // [N][64]
  __shared__ float sC[NODES_PER_BLOCK][2 * DFEAT + 1]; // +1 pad: bank-conflict break
  __shared__ float sW[2 * DFEAT][DOUT + 1];

  const int tid = threadIdx.x;
  const long long node0 = (long long)blockIdx.x * NODES_PER_BLOCK;

  // Stage W: 8192 elems / 256 threads = 32 each (coalesced rows of 64).
  for (int i = tid; i < 2 * DFEAT * DOUT; i += 256) {
    sW[i >> 6][i & 63] = W[i];
  }
  // Stage combined tile: cols [0,64) = x row, cols [64,128) = neigh mean.
  for (int i = tid; i < NODES_PER_BLOCK * 2 * DFEAT; i += 256) {
    int r = i >> 7;
    int c = i & 127;
    long long node = node0 + r;
    float v = 0.0f;
    if (node < NNODES) {
      if (c < DFEAT) {
        v = x[node * DFEAT + c];
      } else {
        float dg = deg[node];
        v = neigh_sum[node * DFEAT + (c - DFEAT)] * (1.0f / fmaxf(dg, 1.0f));
      }
    }
    sC[r][c] = v;
  }
  __syncthreads();

  const int wave = tid >> 5;          // 0..7
  const int lane = tid & 31;
  const int mt   = wave >> 2;         // 0..1 : which 16-row group
  const int nt   = wave & 3;          // 0..3 : which 16-col group
  const int lrow = lane & 15;
  const int hi   = lane >> 4;         // 0: lanes 0-15, 1: lanes 16-31
  const int kOff = hi << 1;           // ISA A/B layout: upper half holds K+2,K+3

  const int arow = mt * 16 + lrow;    // A: M = lane%16
  const int bcol = nt * 16 + lrow;    // B: N = lane%16

  v8f acc = {};
#pragma unroll
  for (int k = 0; k < 2 * DFEAT; k += 4) {
    v2f a, bf;
    a.x  = sC[arow][k + kOff];
    a.y  = sC[arow][k + kOff + 1];
    bf.x = sW[k + kOff][bcol];
    bf.y = sW[k + kOff + 1][bcol];
    // (neg_a, A, neg_b, B, c_mod, C, reuse_a, reuse_b)
    acc = __builtin_amdgcn_wmma_f32_16x16x4_f32(false, a, false, bf,
                                                (short)0, acc, false, false);
  }

  const float bb = bias[bcol];
  const int rbase = hi * 8;           // C/D layout: VGPR r -> M = r + 8*(lane>=16)
#pragma unroll
  for (int r = 0; r < 8; ++r) {
    long long node = node0 + mt * 16 + rbase + r;
    if (node < NNODES) out[node * DOUT + bcol] = acc[r] + bb;
  }
}

// ---------------- Launch ---------------------------------------------------
extern "C" void kernel_launch(void* const* d_in, const int* in_sizes, int n_in,
                              void* d_out, int out_size, void* d_ws, size_t ws_size,
                              hipStream_t stream) {
  const float* x        = (const float*)d_in[0];
  const int*   edge_src = (const int*)d_in[1];
  const int*   edge_dst = (const int*)d_in[2];
  const float* W        = (const float*)d_in[3];
  const float* bias     = (const float*)d_in[4];
  float* out = (float*)d_out;

  const long long E = in_sizes[1];

  float* neigh_sum = (float*)d_ws;                       // N*64 floats
  float* deg       = neigh_sum + (long long)NNODES * DFEAT; // N floats

  // 1) zero neigh_sum + deg
  {
    long long n = (long long)NNODES * (DFEAT + 1);
    int blocks = (int)((n + 255) / 256);
    sage_zero_ws<<<blocks, 256, 0, stream>>>(neigh_sum, n);
  }
  // 2) scatter
  {
    long long nwork = E * DFEAT;
    long long b = (nwork + 255) / 256;
    int blocks = (b > 262144) ? 262144 : (int)b;   // grid-stride covers the rest
    sage_scatter<<<blocks, 256, 0, stream>>>(x, edge_src, edge_dst,
                                             neigh_sum, deg, nwork);
  }
  // 3) fused normalize + concat + GEMM + bias
  {
    int blocks = (NNODES + NODES_PER_BLOCK - 1) / NODES_PER_BLOCK; // 3125
    sage_gemm<<<blocks, 256, 0, stream>>>(x, W, bias, neigh_sum, deg, out);
  }
}